// AFNO1DSeq_17961553231982
// MI455X (gfx1250) — compile-verified
//
#include <hip/hip_runtime.h>
#include <math.h>

typedef __attribute__((ext_vector_type(2))) float v2f;
typedef __attribute__((ext_vector_type(8))) float v8f;

#define TT   256
#define CC   256
#define LAM  0.01f

// LDS layout (float offsets). Sized for worst case P'=512 rows of 32 channels.
#define OFF_XRE   0        // 512*32
#define OFF_XIM   16384
#define OFF_O1RE  32768
#define OFF_O1IM  49152
#define OFF_W1R   65536    // 32x32 row-major [k][n]
#define OFF_W1I   66560
#define OFF_W2R   67584
#define OFF_W2I   68608
#define OFF_B1R   69632
#define OFF_B1I   69664
#define OFF_B2R   69696
#define OFF_B2I   69728
#define OFF_PART  69760    // 8*32 partial sums
#define SMEM_FLOATS 70016  // 280064 bytes < 320 KB WGP LDS

static __device__ __forceinline__ v8f wmma4(v2f a, v2f b, v8f c) {
  // D(16x16,f32) += A(16x4,f32) x B(4x16,f32)
  return __builtin_amdgcn_wmma_f32_16x16x4_f32(false, a, false, b, (short)0, c,
                                               false, false);
}

// One complex block-diagonal layer: [P x 32] complex @ [32 x 32] complex (+bias, act)
// A layout in LDS: S[p*32 + k]. W layout: W[k*32 + n] (row-major K x N).
template <bool RELU>
static __device__ __forceinline__ void gemm_layer(
    const float* Are, const float* Aim,
    const float* Wr,  const float* Wi,
    const float* br,  const float* bi,
    float* Or, float* Oi,
    int P, int Mtiles, int lane, int wv) {
  const int nlo = lane & 15;         // column within N-tile (B/C fragment)
  const int kh  = (lane >> 4) << 1;  // K sub-offset per lane half (A/B fragment)
  for (int mt = wv; mt < Mtiles; mt += 8) {
    const int mrow = mt * 16 + (lane & 15);  // A row held by this lane
    const bool inb = (mrow < P);
    v8f RR0 = {0,0,0,0,0,0,0,0};
    v8f RR1 = RR0, II0 = RR0, II1 = RR0, IR0 = RR0, IR1 = RR0, RI0 = RR0, RI1 = RR0;
#pragma unroll
    for (int kk = 0; kk < 8; ++kk) {
      const int k0 = kk * 4 + kh;
      v2f aR, aI;
      aR.x = inb ? Are[mrow * 32 + k0]     : 0.f;
      aR.y = inb ? Are[mrow * 32 + k0 + 1] : 0.f;
      aI.x = inb ? Aim[mrow * 32 + k0]     : 0.f;
      aI.y = inb ? Aim[mrow * 32 + k0 + 1] : 0.f;
      v2f bR0, bR1, bI0, bI1;
      bR0.x = Wr[(k0    ) * 32 + nlo];      bR0.y = Wr[(k0 + 1) * 32 + nlo];
      bR1.x = Wr[(k0    ) * 32 + 16 + nlo]; bR1.y = Wr[(k0 + 1) * 32 + 16 + nlo];
      bI0.x = Wi[(k0    ) * 32 + nlo];      bI0.y = Wi[(k0 + 1) * 32 + nlo];
      bI1.x = Wi[(k0    ) * 32 + 16 + nlo]; bI1.y = Wi[(k0 + 1) * 32 + 16 + nlo];
      RR0 = wmma4(aR, bR0, RR0);  RR1 = wmma4(aR, bR1, RR1);
      II0 = wmma4(aI, bI0, II0);  II1 = wmma4(aI, bI1, II1);
      IR0 = wmma4(aI, bR0, IR0);  IR1 = wmma4(aI, bR1, IR1);
      RI0 = wmma4(aR, bI0, RI0);  RI1 = wmma4(aR, bI1, RI1);
    }
    const float br0 = br[nlo], br1 = br[16 + nlo];
    const float bi0 = bi[nlo], bi1 = bi[16 + nlo];
#pragma unroll
    for (int e = 0; e < 8; ++e) {
      const int row = mt * 16 + ((lane >> 4) << 3) + e;  // C/D: M split by lane half
      float vr0 = RR0[e] - II0[e] + br0;
      float vr1 = RR1[e] - II1[e] + br1;
      float vi0 = IR0[e] + RI0[e] + bi0;
      float vi1 = IR1[e] + RI1[e] + bi1;
      if (RELU) {
        vr0 = fmaxf(vr0, 0.f); vr1 = fmaxf(vr1, 0.f);
        vi0 = fmaxf(vi0, 0.f); vi1 = fmaxf(vi1, 0.f);
      } else {  // softshrink
        vr0 = copysignf(fmaxf(fabsf(vr0) - LAM, 0.f), vr0);
        vr1 = copysignf(fmaxf(fabsf(vr1) - LAM, 0.f), vr1);
        vi0 = copysignf(fmaxf(fabsf(vi0) - LAM, 0.f), vi0);
        vi1 = copysignf(fmaxf(fabsf(vi1) - LAM, 0.f), vi1);
      }
      Or[row * 32 + nlo]      = vr0;
      Or[row * 32 + 16 + nlo] = vr1;
      Oi[row * 32 + nlo]      = vi0;
      Oi[row * 32 + 16 + nlo] = vi1;
    }
  }
}

__global__ __launch_bounds__(256, 1)
void afno1d_kernel(const float* __restrict__ x,  const float* __restrict__ w1,
                   const float* __restrict__ b1, const float* __restrict__ w2,
                   const float* __restrict__ b2, float* __restrict__ out) {
  extern __shared__ float sm[];
  const int tid = threadIdx.x;
  const int bx  = blockIdx.x;
  const int t   = bx >> 5;        // 0..255
  const int b   = (bx >> 3) & 3;  // 0..3
  const int n   = bx & 7;         // channel block 0..7
  const int c0  = n * 32;

  // P = max(8, 1 << bit_length(t+1))  (smallest pow2 strictly > t+1)
  const int v  = t + 1;
  int P = 1 << (32 - __clz(v));
  if (P < 8) P = 8;
  const int lp = 31 - __clz(P);
  const float inv_sqrtP = 1.0f / sqrtf((float)P);

  float* Xre  = sm + OFF_XRE;
  float* Xim  = sm + OFF_XIM;
  float* O1re = sm + OFF_O1RE;
  float* O1im = sm + OFF_O1IM;

  // --- stage this block's weights/biases into LDS ---
  for (int idx = tid; idx < 4096; idx += 256) {
    const int q = idx >> 10, r = idx & 1023;
    const float* src = (q == 0) ? (w1 + (size_t)n * 1024)
                     : (q == 1) ? (w1 + (size_t)(8 + n) * 1024)
                     : (q == 2) ? (w2 + (size_t)n * 1024)
                                : (w2 + (size_t)(8 + n) * 1024);
    sm[OFF_W1R + idx] = src[r];
  }
  for (int idx = tid; idx < 128; idx += 256) {
    const int q = idx >> 5, r = idx & 31;
    const float* src = (q == 0) ? (b1 + n * 32) : (q == 1) ? (b1 + (8 + n) * 32)
                     : (q == 2) ? (b2 + n * 32) :            (b2 + (8 + n) * 32);
    sm[OFF_B1R + idx] = src[r];
  }

  // --- load causal prefix, bit-reversed placement, fold ortho 1/sqrt(P) ---
  const float* xb = x + (size_t)b * TT * CC;
  for (int idx = tid; idx < P * 32; idx += 256) {
    const int p = idx >> 5, c = idx & 31;
    const float val = (p <= t) ? xb[(size_t)p * CC + c0 + c] * inv_sqrtP : 0.f;
    const unsigned rp = __brev((unsigned)p) >> (32 - lp);
    Xre[rp * 32 + c] = val;
    Xim[rp * 32 + c] = 0.f;
  }
  __syncthreads();

  // --- radix-2 DIT forward FFT (e^{-i...}), in-place in LDS ---
  for (int s = 1; s <= lp; ++s) {
    const int len = 1 << s, half = len >> 1;
    for (int idx = tid; idx < (P >> 1) * 32; idx += 256) {
      const int c = idx & 31, bf = idx >> 5;
      const int j = bf & (half - 1);
      const int g = bf >> (s - 1);
      const int k = g * len + j;
      const float ang = -6.28318530717958648f * (float)j / (float)len;
      float sv, cv;
      __sincosf(ang, &sv, &cv);
      const int i0 = k * 32 + c, i1 = (k + half) * 32 + c;
      const float ur = Xre[i0], ui = Xim[i0];
      const float tr = Xre[i1], ti = Xim[i1];
      const float wr = tr * cv - ti * sv;
      const float wi = tr * sv + ti * cv;
      Xre[i0] = ur + wr; Xim[i0] = ui + wi;
      Xre[i1] = ur - wr; Xim[i1] = ui - wi;
    }
    __syncthreads();
  }

  const int lane = tid & 31, wv = tid >> 5;
  const int Mtiles = ((P < 16) ? 16 : P) >> 4;

  // layer 1: relu(complex matmul + bias) -> O1
  gemm_layer<true>(Xre, Xim, sm + OFF_W1R, sm + OFF_W1I,
                   sm + OFF_B1R, sm + OFF_B1I, O1re, O1im, P, Mtiles, lane, wv);
  __syncthreads();
  // layer 2: softshrink(complex matmul + bias) -> back into X buffers (o2)
  gemm_layer<false>(O1re, O1im, sm + OFF_W2R, sm + OFF_W2I,
                    sm + OFF_B2R, sm + OFF_B2I, Xre, Xim, P, Mtiles, lane, wv);
  __syncthreads();

  // --- inverse ortho DFT evaluated only at point t: Re{ sum_p o2[p] e^{+2πi p t / P} } / sqrt(P) ---
  {
    const int c = lane, chunk = wv;
    float acc = 0.f;
    for (int p = chunk; p < P; p += 8) {
      const int ph = (p * t) & (P - 1);  // exact integer phase mod P
      const float ang = 6.28318530717958648f * (float)ph / (float)P;
      float sv, cv;
      __sincosf(ang, &sv, &cv);
      acc += Xre[p * 32 + c] * cv - Xim[p * 32 + c] * sv;
    }
    sm[OFF_PART + chunk * 32 + c] = acc;
  }
  __syncthreads();
  if (tid < 32) {
    float tot = 0.f;
#pragma unroll
    for (int k = 0; k < 8; ++k) tot += sm[OFF_PART + k * 32 + tid];
    tot *= inv_sqrtP;
    const size_t gi = (size_t)b * TT * CC + (size_t)t * CC + c0 + tid;
    out[gi] = tot + x[gi];  // residual
  }
}

extern "C" void kernel_launch(void* const* d_in, const int* in_sizes, int n_in,
                              void* d_out, int out_size, void* d_ws, size_t ws_size,
                              hipStream_t stream) {
  (void)in_sizes; (void)n_in; (void)out_size; (void)d_ws; (void)ws_size;
  const float* x  = (const float*)d_in[0];
  const float* w1 = (const float*)d_in[1];
  const float* b1 = (const float*)d_in[2];
  const float* w2 = (const float*)d_in[3];
  const float* b2 = (const float*)d_in[4];
  float* out = (float*)d_out;

  dim3 grid(TT * 4 * 8);  // (t, batch, channel-block)
  dim3 block(256);        // 8 wave32
  const size_t smem = SMEM_FLOATS * sizeof(float);
  hipLaunchKernelGGL(afno1d_kernel, grid, block, smem, stream, x, w1, b1, w2, b2, out);
}